// Network_27599459844593
// MI455X (gfx1250) — compile-verified
//
#include <hip/hip_runtime.h>

typedef float v2f __attribute__((ext_vector_type(2)));
typedef float v8f __attribute__((ext_vector_type(8)));

#define NN 100000
#define NE 1600000

// ---------------------------------------------------------------------------
// Dense GEMM (C = A[M,K] * B[K,N] + bias, optional ReLU on A) using
// V_WMMA_F32_16X16X4_F32. One wave owns one 16x16 output tile.
//   A layout (16x4 f32): lane 0-15 row M=lane&15, v[0]=K+0, v[1]=K+1;
//                        lanes 16-31 same rows, v[0]=K+2, v[1]=K+3.
//   B layout (4x16 f32): v[0] lanes0-15 row K+0 / lanes16-31 row K+2;
//                        v[1] row K+1 / row K+3; col = n0 + (lane&15).
//   C layout: VGPR r -> row m0 + 8*(lane>>4) + r, col n0 + (lane&15).
// M (=100000) is an exact multiple of 16 -> no bounds checks, EXEC all-1s.
// ---------------------------------------------------------------------------
template<int K_, int N_, bool RELU>
__global__ __launch_bounds__(256)
void gemm_bias_wmma(const float* __restrict__ A, const float* __restrict__ B,
                    const float* __restrict__ bias, float* __restrict__ C) {
  const int lane = threadIdx.x & 31;
  const int wid  = threadIdx.x >> 5;
  const int m0   = blockIdx.x * 16;
  const int n0   = wid * 16;          // one 16-wide N tile per wave
  const int half = lane >> 4;         // 0: K+0/K+1, 1: K+2/K+3
  const int l15  = lane & 15;

  const float* __restrict__ arow = A + (size_t)(m0 + l15) * K_;
  const float* __restrict__ bcol = B + n0 + l15;

  v8f acc = {};
#pragma unroll
  for (int k = 0; k < K_; k += 4) {
    const int ka = k + 2 * half;
    float a0 = arow[ka];
    float a1 = arow[ka + 1];
    if (RELU) { a0 = fmaxf(a0, 0.0f); a1 = fmaxf(a1, 0.0f); }
    v2f a = { a0, a1 };
    v2f b = { bcol[(size_t)ka * N_], bcol[(size_t)(ka + 1) * N_] };
    // 8 args: (neg_a, A, neg_b, B, c_mod, C, reuse_a, reuse_b)
    acc = __builtin_amdgcn_wmma_f32_16x16x4_f32(
        false, a, false, b, (short)0, acc, false, false);
  }

  const float bn = bias[n0 + l15];
  float* __restrict__ crow = C + (size_t)(m0 + half * 8) * N_ + n0 + l15;
#pragma unroll
  for (int r = 0; r < 8; ++r)
    crow[(size_t)r * N_] = acc[r] + bn;
}

// ---------------------------------------------------------------------------
// COO SPMM scatter: out[rows[e]] += vals[e] * Z[cols[e]].
// One wave handles (32 >> LOGC) edges; each lane owns a float4 chunk, so a
// full feature row is a single coalesced 512B/256B L2 transaction and the 4
// fp32 atomics per lane hit the same cachelines (L2-resident: h/agg < 192MB).
// LOGC: chunks-of-4-floats per edge = 1<<LOGC; D = 4<<LOGC.
// ---------------------------------------------------------------------------
template<int LOGC>
__global__ __launch_bounds__(256)
void spmm_scatter(const int* __restrict__ rows, const int* __restrict__ cols,
                  const float* __restrict__ vals, const float* __restrict__ Z,
                  float* __restrict__ out) {
  const int D = 4 << LOGC;
  const long long t = (long long)blockIdx.x * blockDim.x + threadIdx.x;
  const int e = (int)(t >> LOGC);
  if (e >= NE) return;
  const int c = ((int)t & ((1 << LOGC) - 1)) * 4;

  const int   col = cols[e];
  const int   row = rows[e];
  const float v   = vals[e];

  const float4 z = *reinterpret_cast<const float4*>(Z + (size_t)col * D + c);
  float* o = out + (size_t)row * D + c;
  atomicAdd(o + 0, v * z.x);   // non-returning -> global_atomic_add_f32
  atomicAdd(o + 1, v * z.y);
  atomicAdd(o + 2, v * z.z);
  atomicAdd(o + 3, v * z.w);
}

// Graph-capture-safe zero fill (no hipMemset*).
__global__ __launch_bounds__(256)
void fill_zero4(float4* __restrict__ p, long long n4) {
  const long long i = (long long)blockIdx.x * blockDim.x + threadIdx.x;
  if (i < n4) p[i] = make_float4(0.f, 0.f, 0.f, 0.f);
}

extern "C" void kernel_launch(void* const* d_in, const int* in_sizes, int n_in,
                              void* d_out, int out_size, void* d_ws, size_t ws_size,
                              hipStream_t stream) {
  (void)in_sizes; (void)n_in; (void)out_size; (void)ws_size;

  const int*   rows = (const int*)  d_in[0];
  const int*   cols = (const int*)  d_in[1];
  const float* vals = (const float*)d_in[2];
  const float* x    = (const float*)d_in[3];
  const float* W1   = (const float*)d_in[4];
  const float* b1   = (const float*)d_in[5];
  const float* W2   = (const float*)d_in[6];
  const float* b2   = (const float*)d_in[7];
  float* out = (float*)d_out;

  // Workspace: h1[100000x128] | agg1[100000x128] | h2[100000x64]  (~128 MB)
  float* h1   = (float*)d_ws;
  float* agg1 = h1   + (size_t)NN * 128;
  float* h2   = agg1 + (size_t)NN * 128;

  // Layer 1 GEMM: h1 = x @ W1 + b1        (6250 blocks x 8 waves)
  gemm_bias_wmma<128, 128, false><<<NN / 16, 256, 0, stream>>>(x, W1, b1, h1);

  // agg1 = 0
  {
    long long n4 = (long long)NN * 128 / 4;
    fill_zero4<<<(unsigned)((n4 + 255) / 256), 256, 0, stream>>>((float4*)agg1, n4);
  }
  // agg1 += spmm(h1)   : 1.6M edges x 32 lanes
  {
    long long T = (long long)NE << 5;
    spmm_scatter<5><<<(unsigned)((T + 255) / 256), 256, 0, stream>>>(
        rows, cols, vals, h1, agg1);
  }

  // Layer 2 GEMM with fused ReLU on A: h2 = relu(agg1) @ W2 + b2
  gemm_bias_wmma<128, 64, true><<<NN / 16, 128, 0, stream>>>(agg1, W2, b2, h2);

  // out = 0
  {
    long long n4 = (long long)NN * 64 / 4;
    fill_zero4<<<(unsigned)((n4 + 255) / 256), 256, 0, stream>>>((float4*)out, n4);
  }
  // out += spmm(h2)    : 1.6M edges x 16 lanes
  {
    long long T = (long long)NE << 4;
    spmm_scatter<4><<<(unsigned)((T + 255) / 256), 256, 0, stream>>>(
        rows, cols, vals, h2, out);
  }
}